// TestMoE3d_75849122448010
// MI455X (gfx1250) — compile-verified
//
#include <hip/hip_runtime.h>

// ---------------------------------------------------------------------------
// MoE grouped GEMM for MI455X (gfx1250, wave32, WMMA + async LDS copies).
//   Y[t, o] = sum_k X[t, k] * W[e(t), o, k]   (f32 throughout)
// M = 16384 (ragged over 8 experts, ranges aligned to 512), K = N = 2048.
// Compute: V_WMMA_F32_16X16X4_F32 (native full-precision matrix op).
// Data movement: double-buffered GLOBAL_LOAD_ASYNC_TO_LDS_B128 (ASYNCcnt),
// with a synchronous fallback if the toolchain lacks the async builtins.
// ---------------------------------------------------------------------------

typedef __attribute__((ext_vector_type(2))) float v2f;
typedef __attribute__((ext_vector_type(4))) float v4f;
typedef __attribute__((ext_vector_type(8))) float v8f;
typedef int v4i __attribute__((vector_size(16)));   // matches builtin proto

constexpr int IN   = 2048;  // K
constexpr int OUT  = 2048;  // N
constexpr int NEXP = 8;

constexpr int BM  = 128;    // block tile rows
constexpr int BN  = 128;    // block tile cols
constexpr int BK  = 32;     // K slab
constexpr int LDT = BK + 4; // padded LDS row stride (36 floats):
                            //  - 16B alignment for b128 stores (36 % 4 == 0)
                            //  - 36*m mod 64 => b64 frag reads hit all 64 banks

#if defined(__AMDGCN__) && __has_builtin(__builtin_amdgcn_global_load_async_to_lds_b128)
#define HAVE_ASYNC_LDS 1
#else
#define HAVE_ASYNC_LDS 0
#endif

// 16B global -> LDS copy. Async path bypasses VGPRs and is tracked by ASYNCcnt.
__device__ __forceinline__ void copy16_g2l(float* lds_dst, const float* gsrc) {
#if HAVE_ASYNC_LDS
    __builtin_amdgcn_global_load_async_to_lds_b128(
        (__attribute__((address_space(1))) v4i*)(v4i*)(float*)gsrc,
        (__attribute__((address_space(3))) v4i*)(v4i*)lds_dst,
        /*imm offset=*/0, /*cpol=*/0);
#else
    *(v4f*)lds_dst = *(const v4f*)gsrc;
#endif
}

template <int N>
__device__ __forceinline__ void wait_asynccnt_le() {
#if HAVE_ASYNC_LDS
#if __has_builtin(__builtin_amdgcn_s_wait_asynccnt)
    __builtin_amdgcn_s_wait_asynccnt(N);
#else
    asm volatile("s_wait_asynccnt %0" :: "i"(N) : "memory");
#endif
#endif
}

__global__ __launch_bounds__(256) void moe_grouped_gemm_f32(
    const float* __restrict__ X,     // [total_tokens][IN]
    const float* __restrict__ W,     // [NEXP][OUT][IN]
    const int*   __restrict__ esz,   // [NEXP] expert sizes
    float*       __restrict__ Y)     // [total_tokens][OUT]
{
    __shared__ float sA[2][BM * LDT];   // X tiles,  [m][k], double buffered
    __shared__ float sB[2][BN * LDT];   // W tiles,  [n][k] (row-major => B^T)

    const int tid  = threadIdx.x;
    const int lane = tid & 31;
    const int wave = tid >> 5;       // 0..7 (8 wave32s per WG)
    const int hl   = lane >> 4;      // lane half: 0 => K{0,1}, 1 => K{2,3}
    const int l15  = lane & 15;

    // 2 (M) x 4 (N) wave grid: each wave computes 64x32 of C = 4x2 fragments.
    const int wm = (wave & 1) * 64;
    const int wn = (wave >> 1) * 32;

    const long row0 = (long)blockIdx.x * BM;   // token row of this tile
    const long col0 = (long)blockIdx.y * BN;   // output col of this tile

    // Expert lookup: ranges are multiples of 512 and BM=128 divides 512, so
    // the whole tile belongs to one expert. Uniform across the block.
    int e = 0;
    {
        long off = 0;
        for (int i = 0; i < NEXP; ++i) {
            long n = (long)esz[i];
            if (row0 < off + n) { e = i; break; }
            off += n;
        }
    }
    const float* We = W + (long)e * OUT * IN;

    // Per-thread staging map: A/B tile = 1024 float4 each, 4 per thread each.
    // Consecutive tids walk consecutive 16B chunks within a row (coalesced).
    const auto issue_slab = [&](int buf, int k0) {
        #pragma unroll
        for (int j = 0; j < 4; ++j) {
            const int idx = tid + j * 256;
            const int r = idx >> 3;          // 0..127
            const int c = (idx & 7) << 2;    // 0,4,...,28
            copy16_g2l(&sA[buf][r * LDT + c], X  + (row0 + r) * (long)IN + k0 + c);
            copy16_g2l(&sB[buf][r * LDT + c], We + (col0 + r) * (long)IN + k0 + c);
        }
    };

    const v8f zero = {};
    v8f acc[4][2];
    #pragma unroll
    for (int mf = 0; mf < 4; ++mf)
        #pragma unroll
        for (int nf = 0; nf < 2; ++nf)
            acc[mf][nf] = zero;

    issue_slab(0, 0);   // prologue: fill buffer 0

    for (int k0 = 0; k0 < IN; k0 += BK) {
        const int  cur  = (k0 / BK) & 1;
        const bool more = (k0 + BK) < IN;

        // Overlap: issue next slab's 8 async copies, then wait until only
        // those 8 are outstanding => current slab is resident (in-order cnt).
        if (more) { issue_slab(cur ^ 1, k0 + BK); wait_asynccnt_le<8>(); }
        else      { wait_asynccnt_le<0>(); }
        __syncthreads();

        // ---- compute: 8 k-slices x (4x2) fragments = 64 WMMAs per slab ----
        #pragma unroll
        for (int kk = 0; kk < BK; kk += 4) {
            const int ko = kk + (hl << 1); // lanes 0-15: K{kk,kk+1}; 16-31: K{kk+2,kk+3}
            v2f a[4], b[2];
            #pragma unroll
            for (int mf = 0; mf < 4; ++mf)
                a[mf] = *(const v2f*)(&sA[cur][(wm + mf * 16 + l15) * LDT + ko]);
            #pragma unroll
            for (int nf = 0; nf < 2; ++nf)
                b[nf] = *(const v2f*)(&sB[cur][(wn + nf * 16 + l15) * LDT + ko]);
            #pragma unroll
            for (int mf = 0; mf < 4; ++mf)
                #pragma unroll
                for (int nf = 0; nf < 2; ++nf)
                    acc[mf][nf] = __builtin_amdgcn_wmma_f32_16x16x4_f32(
                        /*neg_a=*/false, a[mf],
                        /*neg_b=*/false, b[nf],
                        /*c_mod=*/(short)0, acc[mf][nf],
                        /*reuse_a=*/false, /*reuse_b=*/false);
        }
        __syncthreads();   // all waves done reading `cur` before it refills
    }

    // ---- write back: C/D layout => VGPR i holds M = mbase+i (halves +8) ----
    #pragma unroll
    for (int mf = 0; mf < 4; ++mf) {
        #pragma unroll
        for (int nf = 0; nf < 2; ++nf) {
            const long mbase = row0 + wm + mf * 16 + hl * 8;
            const long n     = col0 + wn + nf * 16 + l15;
            #pragma unroll
            for (int i = 0; i < 8; ++i)
                Y[(mbase + i) * (long)OUT + n] = acc[mf][nf][i];
        }
    }
}

extern "C" void kernel_launch(void* const* d_in, const int* in_sizes, int n_in,
                              void* d_out, int out_size, void* d_ws, size_t ws_size,
                              hipStream_t stream) {
    const float* X   = (const float*)d_in[0];  // [total, 2048] f32
    const float* W   = (const float*)d_in[1];  // [8, 2048, 2048] f32
    const int*   esz = (const int*)d_in[2];    // [8] int32
    float*       Y   = (float*)d_out;          // [total, 2048] f32

    const int total = in_sizes[0] / IN;        // 16384
    dim3 grid(total / BM, OUT / BN);           // (128, 16)
    moe_grouped_gemm_f32<<<grid, 256, 0, stream>>>(X, W, esz, Y);
}